// QReLU_22359599743361
// MI455X (gfx1250) — compile-verified
//
#include <hip/hip_runtime.h>
#include <hip/hip_bf16.h>

// QReLU: out = where(x > 0, x, -1.99*x)  ==  max(x, -1.99f * x)
//
// Memory-bound streaming kernel: 1 GB in + 1 GB out @ 23.3 TB/s -> ~92 us floor.
// Strategy:
//   - b128 vector loads/stores (16 B/lane -> one 512 B line per wave32 access)
//   - non-temporal hints: 2 GB stream >> 192 MB L2, data touched exactly once
//   - checkless full-tile kernel: 4 NT loads issued back-to-back for
//     intra-thread MLP, then compute, then 4 NT stores (no branches, no exec
//     masking in the hot path)
//   - branch-free math: max(x, -1.99x) (sign analysis covers all cases,
//     -0 -> +0 matches IEEE maxNum, NaN propagates since both operands NaN)

typedef float v4f __attribute__((ext_vector_type(4)));

constexpr int TPB = 256;   // 8 wave32s per block
constexpr int VPT = 4;     // float4s per thread (64 B per thread)

__device__ __forceinline__ v4f qrelu4(v4f v) {
    v4f r;
    r.x = fmaxf(v.x, -1.99f * v.x);
    r.y = fmaxf(v.y, -1.99f * v.y);
    r.z = fmaxf(v.z, -1.99f * v.z);
    r.w = fmaxf(v.w, -1.99f * v.w);
    return r;
}

// Full-tile kernel: every block covers exactly TPB*VPT float4s, no bounds
// checks. Threads within each of the VPT chunks are contiguous, so every
// b128 access is perfectly coalesced.
__global__ __launch_bounds__(TPB) void qrelu_main_kernel(const v4f* __restrict__ in,
                                                         v4f* __restrict__ out) {
    long long base = (long long)blockIdx.x * (TPB * VPT) + threadIdx.x;

    // Issue all loads first -> 4 NT b128 loads in flight per thread.
    v4f a = __builtin_nontemporal_load(&in[base + 0 * TPB]);
    v4f b = __builtin_nontemporal_load(&in[base + 1 * TPB]);
    v4f c = __builtin_nontemporal_load(&in[base + 2 * TPB]);
    v4f d = __builtin_nontemporal_load(&in[base + 3 * TPB]);

    a = qrelu4(a);
    b = qrelu4(b);
    c = qrelu4(c);
    d = qrelu4(d);

    __builtin_nontemporal_store(a, &out[base + 0 * TPB]);
    __builtin_nontemporal_store(b, &out[base + 1 * TPB]);
    __builtin_nontemporal_store(c, &out[base + 2 * TPB]);
    __builtin_nontemporal_store(d, &out[base + 3 * TPB]);
}

// Scalar tail: covers elements [start, n). Not hit for this shape
// (n = 2^28 -> exactly 65,536 full blocks), kept for generality.
__global__ __launch_bounds__(TPB) void qrelu_tail_kernel(const float* __restrict__ in,
                                                         float* __restrict__ out,
                                                         long long start, long long n) {
    long long i = start + (long long)blockIdx.x * blockDim.x + threadIdx.x;
    if (i < n) {
        float x = in[i];
        out[i] = fmaxf(x, -1.99f * x);
    }
}

extern "C" void kernel_launch(void* const* d_in, const int* in_sizes, int n_in,
                              void* d_out, int out_size, void* d_ws, size_t ws_size,
                              hipStream_t stream) {
    (void)n_in; (void)d_ws; (void)ws_size; (void)out_size;

    const float* x = (const float*)d_in[0];
    float* y = (float*)d_out;
    long long n = (long long)in_sizes[0];   // 32*4096*2048 = 268,435,456

    long long n4 = n >> 2;                              // float4 count
    constexpr long long F4_PER_BLOCK = (long long)TPB * VPT;  // 1024
    long long full_blocks = n4 / F4_PER_BLOCK;          // 65,536 for this shape

    if (full_blocks > 0) {
        qrelu_main_kernel<<<(int)full_blocks, TPB, 0, stream>>>(
            (const v4f*)x, (v4f*)y);
    }

    long long covered = full_blocks * F4_PER_BLOCK * 4; // elements handled above
    long long tail = n - covered;
    if (tail > 0) {
        int tblocks = (int)((tail + TPB - 1) / TPB);
        qrelu_tail_kernel<<<tblocks, TPB, 0, stream>>>(x, y, covered, n);
    }
}